// ResGatedGraphConv_10995116277968
// MI455X (gfx1250) — compile-verified
//
#include <hip/hip_runtime.h>

typedef __attribute__((ext_vector_type(2))) float v2f;
typedef __attribute__((ext_vector_type(8))) float v8f;

#define N_NODES 100000
#define N_EDGES 640000
#define NIN 128
#define NOUT 128
#define LDS_STRIDE 132   // 128 + 4 floats pad -> rows land on distinct banks

// ---------------------------------------------------------------------------
// Kernel 1: K/Q/V/Skip GEMMs via V_WMMA_F32_16X16X4_F32 (fp32, no downcast).
// Block = 256 threads = 8 waves. blockIdx.x = 16-row tile (100000/16 = 6250).
// Each wave owns 4 N-tiles (64 cols) of ONE weight matrix:
//   matrix = wave>>1 (0=k,1=q,2=v,3=skip), column base = (wave&1)*64.
// The A-fragment (shared across N) is loaded once per k-step and feeds 4
// WMMAs, amortizing LDS reads and loop overhead 4x vs one-tile-per-wave.
// Skip result (+bias) is written directly to d_out; edge kernel adds on top.
// ---------------------------------------------------------------------------
__global__ __launch_bounds__(256)
void gemm_kqvs(const float* __restrict__ x,
               const float* __restrict__ Wk, const float* __restrict__ bk,
               const float* __restrict__ Wq, const float* __restrict__ bq,
               const float* __restrict__ Wv, const float* __restrict__ bv,
               const float* __restrict__ Ws, const float* __restrict__ bs,
               float* __restrict__ Kout, float* __restrict__ Qout,
               float* __restrict__ Vout, float* __restrict__ Sout)
{
    __shared__ float xs[16 * LDS_STRIDE];

    const int tid  = threadIdx.x;
    const int wave = tid >> 5;
    const int lane = tid & 31;
    const int m0   = blockIdx.x * 16;

    // Cooperative load of the 16x128 x-tile into padded LDS.
    {
        const int row = tid >> 4;         // 0..15
        const int col = (tid & 15) * 8;   // 0,8,...,120
        const float4* s = (const float4*)(x + (size_t)(m0 + row) * NIN + col);
        float4 a = s[0];
        float4 b = s[1];
        float* d = xs + row * LDS_STRIDE + col;
        d[0] = a.x; d[1] = a.y; d[2] = a.z; d[3] = a.w;
        d[4] = b.x; d[5] = b.y; d[6] = b.z; d[7] = b.w;
    }
    __syncthreads();

    const int mat   = wave >> 1;          // 0=k 1=q 2=v 3=skip
    const int nbase = (wave & 1) * 64;    // 64-column slab per wave

    const float* W; const float* B; float* O;
    switch (mat) {
        case 0:  W = Wk; B = bk; O = Kout; break;
        case 1:  W = Wq; B = bq; O = Qout; break;
        case 2:  W = Wv; B = bv; O = Vout; break;
        default: W = Ws; B = bs; O = Sout; break;
    }

    const int lhalf = lane >> 4;   // 0: lanes 0-15, 1: lanes 16-31
    const int l15   = lane & 15;

    v8f acc0 = {}, acc1 = {}, acc2 = {}, acc3 = {};

    #pragma unroll 2
    for (int k0 = 0; k0 < NIN; k0 += 4) {
        const int kk = k0 + 2 * lhalf;
        // A frag (16x4 f32): lane half selects K=k0..k0+1 vs K=k0+2..k0+3
        v2f a;
        a.x = xs[l15 * LDS_STRIDE + kk];
        a.y = xs[l15 * LDS_STRIDE + kk + 1];

        // B frags (4x16 f32) for the 4 N-tiles: row K=kk / K=kk+1, col n0+l15
        const float* w0 = W + (size_t)kk * NOUT + nbase + l15;
        const float* w1 = W + (size_t)(kk + 1) * NOUT + nbase + l15;
        v2f b0, b1, b2, b3;
        b0.x = w0[0];  b0.y = w1[0];
        b1.x = w0[16]; b1.y = w1[16];
        b2.x = w0[32]; b2.y = w1[32];
        b3.x = w0[48]; b3.y = w1[48];

        acc0 = __builtin_amdgcn_wmma_f32_16x16x4_f32(
                   false, a, false, b0, (short)0, acc0, false, false);
        acc1 = __builtin_amdgcn_wmma_f32_16x16x4_f32(
                   false, a, false, b1, (short)0, acc1, false, false);
        acc2 = __builtin_amdgcn_wmma_f32_16x16x4_f32(
                   false, a, false, b2, (short)0, acc2, false, false);
        acc3 = __builtin_amdgcn_wmma_f32_16x16x4_f32(
                   false, a, false, b3, (short)0, acc3, false, false);
    }

    // C/D layout: VGPR r -> row r (lanes 0-15) / row 8+r (lanes 16-31), col = lane&15
    const float bv0 = B[nbase + l15];
    const float bv1 = B[nbase + 16 + l15];
    const float bv2 = B[nbase + 32 + l15];
    const float bv3 = B[nbase + 48 + l15];

    #pragma unroll
    for (int r = 0; r < 8; ++r) {
        const int row = m0 + r + 8 * lhalf;
        float* o = O + (size_t)row * NOUT + nbase + l15;
        o[0]  = acc0[r] + bv0;
        o[16] = acc1[r] + bv1;
        o[32] = acc2[r] + bv2;
        o[48] = acc3[r] + bv3;
    }
}

// ---------------------------------------------------------------------------
// Kernel 2: per-edge gate + scatter-add. One wave per edge; lane -> float4 of
// the 128 features (coalesced 512B gathers, served mostly from 192MB L2).
// ---------------------------------------------------------------------------
__global__ __launch_bounds__(256)
void edge_gate_scatter(const long long* __restrict__ ei,
                       const float* __restrict__ K,
                       const float* __restrict__ Q,
                       const float* __restrict__ V,
                       float* __restrict__ out)
{
    const int wave = threadIdx.x >> 5;
    const int lane = threadIdx.x & 31;
    const long long e = (long long)blockIdx.x * 8 + wave;
    if (e >= N_EDGES) return;

    const int src = (int)ei[e];
    const int dst = (int)ei[(size_t)N_EDGES + e];

    const float4 kd = ((const float4*)(K + (size_t)dst * NOUT))[lane];
    const float4 qs = ((const float4*)(Q + (size_t)src * NOUT))[lane];
    const float4 vs = ((const float4*)(V + (size_t)src * NOUT))[lane];

    float4 m;
    m.x = vs.x / (1.0f + __expf(-(kd.x + qs.x)));
    m.y = vs.y / (1.0f + __expf(-(kd.y + qs.y)));
    m.z = vs.z / (1.0f + __expf(-(kd.z + qs.z)));
    m.w = vs.w / (1.0f + __expf(-(kd.w + qs.w)));

    float* o = out + (size_t)dst * NOUT + lane * 4;
    __hip_atomic_fetch_add(o + 0, m.x, __ATOMIC_RELAXED, __HIP_MEMORY_SCOPE_AGENT);
    __hip_atomic_fetch_add(o + 1, m.y, __ATOMIC_RELAXED, __HIP_MEMORY_SCOPE_AGENT);
    __hip_atomic_fetch_add(o + 2, m.z, __ATOMIC_RELAXED, __HIP_MEMORY_SCOPE_AGENT);
    __hip_atomic_fetch_add(o + 3, m.w, __ATOMIC_RELAXED, __HIP_MEMORY_SCOPE_AGENT);
}

// ---------------------------------------------------------------------------
// Inputs (setup_inputs order):
//  0:x 1:edge_index(int64) 2:edge_attr(unused) 3:Wk 4:bk 5:Wq 6:bq 7:Wv 8:bv
//  9:Wskip 10:bias
// Workspace: 3 * 100000 * 128 * 4 B = 153.6 MB for K,Q,V.
// ---------------------------------------------------------------------------
extern "C" void kernel_launch(void* const* d_in, const int* in_sizes, int n_in,
                              void* d_out, int out_size, void* d_ws, size_t ws_size,
                              hipStream_t stream) {
    const float*     x  = (const float*)d_in[0];
    const long long* ei = (const long long*)d_in[1];
    const float* Wk = (const float*)d_in[3];
    const float* bk = (const float*)d_in[4];
    const float* Wq = (const float*)d_in[5];
    const float* bq = (const float*)d_in[6];
    const float* Wv = (const float*)d_in[7];
    const float* bv = (const float*)d_in[8];
    const float* Ws = (const float*)d_in[9];
    const float* bs = (const float*)d_in[10];

    float* out = (float*)d_out;
    float* Kb  = (float*)d_ws;
    float* Qb  = Kb + (size_t)N_NODES * NOUT;
    float* Vb  = Qb + (size_t)N_NODES * NOUT;

    gemm_kqvs<<<dim3(N_NODES / 16), 256, 0, stream>>>(
        x, Wk, bk, Wq, bq, Wv, bv, Ws, bs, Kb, Qb, Vb, out);

    edge_gate_scatter<<<N_EDGES / 8, 256, 0, stream>>>(ei, Kb, Qb, Vb, out);
}